// ConvOneByOne_40175124087517
// MI455X (gfx1250) — compile-verified
//
#include <hip/hip_runtime.h>
#include <hip/hip_bf16.h>

typedef __attribute__((ext_vector_type(2))) float v2f;
typedef __attribute__((ext_vector_type(8))) float v8f;

#define NCH  1024
#define ALEN 4096

// out[s,a] = sum_c w[c] * x[s,c,a] + b
// One wave computes 32 consecutive 'a' outputs (two 16-wide WMMA tiles),
// accumulating over all 1024 channels with chained V_WMMA_F32_16X16X4_F32.
//
// A-matrix (16x4 f32, documented layout):
//   lane L<16 : M=L, VGPR0=K0, VGPR1=K1   -> x[s, c+0/1, a0+L]
//   lane 16+L : M=L, VGPR0=K2, VGPR1=K3   -> x[s, c+2/3, a0+L]
// B-matrix (4x16): B[k][n] = w[c+k] for all n (broadcast from LDS), so every
// column of D holds the same per-row result: D[m][*] = sum_k x[.,c+k,a0+m]*w[c+k].
__global__ __launch_bounds__(256) void conv1x1_wmma_kernel(
    const float* __restrict__ x, const float* __restrict__ w,
    const float* __restrict__ bptr, float* __restrict__ out) {
  __shared__ float ws[NCH];
  for (int i = threadIdx.x; i < NCH; i += 256) ws[i] = w[i];
  __syncthreads();

  const int lane = threadIdx.x & 31;
  const int wave = threadIdx.x >> 5;
  const int s    = blockIdx.y;
  const int a0   = blockIdx.x * 256 + wave * 32;  // this wave: a0 .. a0+31
  const int half = lane >> 4;                     // 0: K=0,1  1: K=2,3
  const int sub  = lane & 15;                     // M index within tile
  const int koff = half * 2;

  // Base pointer for this lane's A elements (K offset folded in).
  const float* xp = x + (size_t)s * NCH * ALEN + (size_t)koff * ALEN + a0 + sub;

  v8f acc0 = {};
  v8f acc1 = {};

#pragma unroll 4
  for (int c = 0; c < NCH; c += 4) {
    // B operand: w[c+koff], w[c+koff+1] broadcast across the 16-lane half.
    v2f bv;
    bv.x = ws[c + koff];
    bv.y = ws[c + koff + 1];

    const float* p = xp + (size_t)c * ALEN;
    v2f av0, av1;
    av0.x = __builtin_nontemporal_load(p);               // tile0, K = koff
    av0.y = __builtin_nontemporal_load(p + ALEN);        // tile0, K = koff+1
    av1.x = __builtin_nontemporal_load(p + 16);          // tile1 (a0+16..a0+31)
    av1.y = __builtin_nontemporal_load(p + ALEN + 16);

    acc0 = __builtin_amdgcn_wmma_f32_16x16x4_f32(
        false, av0, false, bv, (short)0, acc0, false, false);
    acc1 = __builtin_amdgcn_wmma_f32_16x16x4_f32(
        false, av1, false, bv, (short)0, acc1, false, false);
  }

  // D layout: component r, lanes 0-15 -> M=r ; lanes 16-31 -> M=8+r.
  // All columns identical, so lane 0 (M=0..7) and lane 16 (M=8..15) write out.
  const float bias = bptr[0];
  if (sub == 0) {
    float* o = out + (size_t)s * ALEN + a0 + half * 8;
    float4 v;
    v = make_float4(acc0[0] + bias, acc0[1] + bias, acc0[2] + bias, acc0[3] + bias);
    *(float4*)(o + 0) = v;
    v = make_float4(acc0[4] + bias, acc0[5] + bias, acc0[6] + bias, acc0[7] + bias);
    *(float4*)(o + 4) = v;
    v = make_float4(acc1[0] + bias, acc1[1] + bias, acc1[2] + bias, acc1[3] + bias);
    *(float4*)(o + 16) = v;
    v = make_float4(acc1[4] + bias, acc1[5] + bias, acc1[6] + bias, acc1[7] + bias);
    *(float4*)(o + 20) = v;
  }
}

extern "C" void kernel_launch(void* const* d_in, const int* in_sizes, int n_in,
                              void* d_out, int out_size, void* d_ws, size_t ws_size,
                              hipStream_t stream) {
  const float* x = (const float*)d_in[0];
  const float* w = (const float*)d_in[1];
  const float* b = (const float*)d_in[2];
  float* out = (float*)d_out;

  const int S = in_sizes[0] / (NCH * ALEN);  // 64 samples
  dim3 grid(ALEN / 256, S);                  // 16 x 64 blocks
  conv1x1_wmma_kernel<<<grid, 256, 0, stream>>>(x, w, b, out);
}